// MultiLayerGNN_22557168239476
// MI455X (gfx1250) — compile-verified
//
#include <hip/hip_runtime.h>
#include <hip/hip_bf16.h>
#include <math.h>

#define N_NODES 100000
#define E_EDGES 800000
#define F_IN    768
#define NHID    128
#define NG      128

#define BM  64      // rows per block tile
#define KB  32      // K chunk
#define KBP 34      // padded LDS stride (even -> 8B-aligned float2, order-32 mod 64 banks)

typedef float v2f __attribute__((ext_vector_type(2)));
typedef float v8f __attribute__((ext_vector_type(8)));

// ---------------- init: zero ACC, DEG, G ----------------
__global__ void gnn_init_kernel(float* __restrict__ acc, float* __restrict__ deg,
                                float* __restrict__ g) {
  size_t stride = (size_t)gridDim.x * blockDim.x;
  size_t tid = (size_t)blockIdx.x * blockDim.x + threadIdx.x;
  for (size_t i = tid; i < (size_t)N_NODES * NHID; i += stride) acc[i] = 0.0f;
  for (size_t i = tid; i < (size_t)N_NODES;        i += stride) deg[i] = 0.0f;
  for (size_t i = tid; i < (size_t)NG * NHID;      i += stride) g[i]   = 0.0f;
}

// ---------------- degree ----------------
__global__ void gnn_degree_kernel(const int* __restrict__ dst, float* __restrict__ deg) {
  int e = blockIdx.x * blockDim.x + threadIdx.x;
  if (e < E_EDGES)
    __hip_atomic_fetch_add(&deg[dst[e]], 1.0f, __ATOMIC_RELAXED, __HIP_MEMORY_SCOPE_AGENT);
}

// ---------------- fused dual-weight fp32 WMMA GEMM ----------------
// Yl = X @ Wl ; Yr = X @ Wr + bias   (Yr may alias X: each block reads all K
// of its own rows before storing)
__global__ __launch_bounds__(256) void gnn_gemm_dual_kernel(
    const float* __restrict__ X, const float* __restrict__ Wl,
    const float* __restrict__ Wr, const float* __restrict__ bias,
    float* __restrict__ Yl, float* __restrict__ Yr, int K) {
  __shared__ float xs[BM * KBP];          //  8.7 KB  x tile (row-major, padded)
  __shared__ float wlt[NHID * KBP];       // 17.4 KB  Wl chunk, transposed [c][k]
  __shared__ float wrt[NHID * KBP];       // 17.4 KB  Wr chunk, transposed [c][k]

  const int tid  = threadIdx.x;
  const int lane = tid & 31;
  const int wave = tid >> 5;
  const int row0 = blockIdx.x * BM;

  const bool isL  = (wave < 4);
  const int rbase = (wave & 3) * 16;            // 16-row strip per wave
  const float* sw = isL ? wlt : wrt;

  v8f acc[8] = {};                              // 8 col-tiles of 16x16 f32

  const int l  = lane & 15;
  const int hi = lane >> 4;                     // K-half select per ISA A/B layout

  for (int kc = 0; kc < K; kc += KB) {
    __syncthreads();
    // stage x tile: 64x32 floats, coalesced global reads
#pragma unroll
    for (int it = 0; it < (BM * KB) / 256; ++it) {
      int idx = tid + it * 256;
      int r = idx >> 5, k = idx & 31;
      int gr = row0 + r;
      xs[r * KBP + k] = (gr < N_NODES) ? X[(size_t)gr * K + kc + k] : 0.0f;
    }
    // stage both weight chunks transposed: wt[c][k] = W[kc+k][c]
#pragma unroll
    for (int it = 0; it < (NHID * KB) / 256; ++it) {
      int idx = tid + it * 256;
      int k = idx >> 7, c = idx & 127;
      wlt[c * KBP + k] = Wl[(size_t)(kc + k) * NHID + c];
      wrt[c * KBP + k] = Wr[(size_t)(kc + k) * NHID + c];
    }
    __syncthreads();

#pragma unroll
    for (int kk = 0; kk < KB; kk += 4) {
      // A frag 16x4: lane<16 -> K={kk,kk+1}, lane>=16 -> K={kk+2,kk+3}
      v2f av = *(const v2f*)&xs[(rbase + l) * KBP + kk + 2 * hi];
#pragma unroll
      for (int t = 0; t < 8; ++t) {
        // B frag 4x16 from transposed LDS: single aligned b64 per lane
        v2f bv = *(const v2f*)&sw[(t * 16 + l) * KBP + kk + 2 * hi];
        acc[t] = __builtin_amdgcn_wmma_f32_16x16x4_f32(
            false, av, false, bv, (short)0, acc[t], false, false);
      }
    }
  }

  float* Y = isL ? Yl : Yr;
#pragma unroll
  for (int t = 0; t < 8; ++t) {
    int col = t * 16 + l;
    float bval = isL ? 0.0f : bias[col];
#pragma unroll
    for (int v = 0; v < 8; ++v) {
      int r = row0 + rbase + v + 8 * hi;        // C/D layout: VGPR v -> M=v / v+8
      if (r < N_NODES) Y[(size_t)r * NHID + col] = acc[t][v] + bval;
    }
  }
}

// ---------------- edge scatter-add (L2 float atomics) ----------------
__global__ void gnn_scatter_kernel(const int* __restrict__ src, const int* __restrict__ dst,
                                   const float* __restrict__ Y, float* __restrict__ acc) {
  size_t tid = (size_t)blockIdx.x * blockDim.x + threadIdx.x;
  size_t e = tid >> 5;
  int c4 = (int)(tid & 31) * 4;
  if (e < E_EDGES) {
    int s = src[e], d = dst[e];
    float4 v = *(const float4*)&Y[(size_t)s * NHID + c4];
    float* a = &acc[(size_t)d * NHID + c4];
    __hip_atomic_fetch_add(a + 0, v.x, __ATOMIC_RELAXED, __HIP_MEMORY_SCOPE_AGENT);
    __hip_atomic_fetch_add(a + 1, v.y, __ATOMIC_RELAXED, __HIP_MEMORY_SCOPE_AGENT);
    __hip_atomic_fetch_add(a + 2, v.z, __ATOMIC_RELAXED, __HIP_MEMORY_SCOPE_AGENT);
    __hip_atomic_fetch_add(a + 3, v.w, __ATOMIC_RELAXED, __HIP_MEMORY_SCOPE_AGENT);
  }
}

// ---------------- combine: H = relu(ACC/max(deg,1) + Yr); optionally re-zero ACC ----
__global__ void gnn_combine_kernel(float* __restrict__ acc, const float* __restrict__ deg,
                                   const float* __restrict__ yr, float* __restrict__ h,
                                   int zeroAcc) {
  size_t tid = (size_t)blockIdx.x * blockDim.x + threadIdx.x;
  size_t i = tid >> 5;
  int c4 = (int)(tid & 31) * 4;
  if (i < N_NODES) {
    float dg = fmaxf(deg[i], 1.0f);
    float inv = 1.0f / dg;
    float4 a = *(const float4*)&acc[i * NHID + c4];
    float4 r = *(const float4*)&yr[i * NHID + c4];
    float4 o;
    o.x = fmaxf(a.x * inv + r.x, 0.0f);
    o.y = fmaxf(a.y * inv + r.y, 0.0f);
    o.z = fmaxf(a.z * inv + r.z, 0.0f);
    o.w = fmaxf(a.w * inv + r.w, 0.0f);
    *(float4*)&h[i * NHID + c4] = o;
    if (zeroAcc) {
      float4 z = {0.f, 0.f, 0.f, 0.f};
      *(float4*)&acc[i * NHID + c4] = z;
    }
  }
}

// ---------------- per-graph max pool (uint-bit atomicMax; inputs >= 0) ----------------
__global__ void gnn_pool_kernel(const float* __restrict__ h, const int* __restrict__ batch,
                                unsigned* __restrict__ g) {
  size_t tid = (size_t)blockIdx.x * blockDim.x + threadIdx.x;
  size_t i = tid >> 5;
  int c4 = (int)(tid & 31) * 4;
  if (i < N_NODES) {
    int b = batch[i];
    float4 v = *(const float4*)&h[i * NHID + c4];
    unsigned* gp = &g[(size_t)b * NHID + c4];
    atomicMax(gp + 0, __float_as_uint(v.x));
    atomicMax(gp + 1, __float_as_uint(v.y));
    atomicMax(gp + 2, __float_as_uint(v.z));
    atomicMax(gp + 3, __float_as_uint(v.w));
  }
}

// ---------------- head: log_softmax(g @ W2 + b2) ----------------
__global__ void gnn_head_kernel(const float* __restrict__ g, const float* __restrict__ W2,
                                const float* __restrict__ b2, float* __restrict__ out) {
  int i = threadIdx.x;
  if (i < NG) {
    float z0 = b2[0], z1 = b2[1];
    for (int c = 0; c < NHID; ++c) {
      float gv = g[i * NHID + c];
      z0 += gv * W2[c * 2 + 0];
      z1 += gv * W2[c * 2 + 1];
    }
    float m = fmaxf(z0, z1);
    float lse = m + logf(expf(z0 - m) + expf(z1 - m));
    out[i * 2 + 0] = z0 - lse;
    out[i * 2 + 1] = z1 - lse;
  }
}

extern "C" void kernel_launch(void* const* d_in, const int* in_sizes, int n_in,
                              void* d_out, int out_size, void* d_ws, size_t ws_size,
                              hipStream_t stream) {
  const float* x   = (const float*)d_in[0];
  const int*   ei  = (const int*)d_in[1];
  const int*   bat = (const int*)d_in[2];
  const float* Wl0 = (const float*)d_in[3];
  const float* bl0 = (const float*)d_in[4];
  const float* Wr0 = (const float*)d_in[5];
  const float* Wl1 = (const float*)d_in[6];
  const float* bl1 = (const float*)d_in[7];
  const float* Wr1 = (const float*)d_in[8];
  const float* W2  = (const float*)d_in[9];
  const float* b2  = (const float*)d_in[10];
  const int* src = ei;
  const int* dst = ei + E_EDGES;

  // workspace carve: 3x 51.2MB + deg + pooled
  float* L   = (float*)d_ws;
  float* R   = L   + (size_t)N_NODES * NHID;
  float* ACC = R   + (size_t)N_NODES * NHID;
  float* DEG = ACC + (size_t)N_NODES * NHID;
  float* G   = DEG + N_NODES;

  const int nGemmBlk = (N_NODES + BM - 1) / BM;
  const int nEdgeVec = (int)(((size_t)E_EDGES * 32 + 255) / 256);
  const int nNodeVec = (int)(((size_t)N_NODES * 32 + 255) / 256);

  gnn_init_kernel<<<2048, 256, 0, stream>>>(ACC, DEG, G);
  gnn_degree_kernel<<<(E_EDGES + 255) / 256, 256, 0, stream>>>(dst, DEG);

  // layer 0: y0l = x@Wl0 -> L ; y0r = x@Wr0 + bl0 -> R
  gnn_gemm_dual_kernel<<<nGemmBlk, 256, 0, stream>>>(x, Wl0, Wr0, bl0, L, R, F_IN);
  gnn_scatter_kernel<<<nEdgeVec, 256, 0, stream>>>(src, dst, L, ACC);
  // h0 = relu(mean + y0r) -> L ; ACC re-zeroed for layer 1
  gnn_combine_kernel<<<nNodeVec, 256, 0, stream>>>(ACC, DEG, R, L, 1);

  // layer 1: y1l = h0@Wl1 -> R ; y1r = h0@Wr1 + bl1 -> L (in-place over h0)
  gnn_gemm_dual_kernel<<<nGemmBlk, 256, 0, stream>>>(L, Wl1, Wr1, bl1, R, L, NHID);
  gnn_scatter_kernel<<<nEdgeVec, 256, 0, stream>>>(src, dst, R, ACC);
  // h1 = relu(mean + y1r) -> R
  gnn_combine_kernel<<<nNodeVec, 256, 0, stream>>>(ACC, DEG, L, R, 0);

  gnn_pool_kernel<<<nNodeVec, 256, 0, stream>>>(R, bat, (unsigned*)G);
  gnn_head_kernel<<<1, 128, 0, stream>>>(G, W2, b2, (float*)d_out);
}